// FFM_7172595384388
// MI455X (gfx1250) — compile-verified
//
#include <hip/hip_runtime.h>
#include <math.h>

// ---------------------------------------------------------------------------
// FFM forward for MI455X (gfx1250):
//   out[b] = sigmoid( x[b]·W + bias + sum_e x_e(b)^T M x_e(b) )
// M[i,j] = dot(V[i,j,:], V[j,i,:]) for i<j else 0   (F=64, E=12, H=4)
//
// Kernel 1: build M directly in WMMA B-fragment order (float2 per lane) so the
//           main kernel's B loads are single ds_load_b64 with immediate offsets.
// Kernel 2: per block = 16 batch rows, 12 waves (one per embedding lane e).
//           x tile staged as e-major LDS planes so A fragments and the fold
//           reads are contiguous b64/b32 with immediate offsets. Linear part
//           fused into the staging pass. V_WMMA_F32_16X16X4_F32 chains.
// ---------------------------------------------------------------------------

typedef float v2f __attribute__((ext_vector_type(2)));
typedef float v8f __attribute__((ext_vector_type(8)));

#define B_TOTAL  32768
#define DIM      768
#define FN       64
#define EM       12
#define ROWS     16
#define XSR      66                 // Xe row stride (floats): 8B-aligned, bank-spread
#define EPLANE   (ROWS * XSR)       // 1056 floats per e-plane
#define NWAVES   12
#define NTHREADS (NWAVES * 32)
#define MPACK_N  (16 * 4 * 32)      // 2048 float2 fragments = 4096 floats

__device__ __forceinline__ float mval(const float* __restrict__ V, int i, int j) {
    if (i >= j) return 0.f;
    const float* a = V + (size_t)(i * FN + j) * 4;
    const float* b = V + (size_t)(j * FN + i) * 4;
    return a[0] * b[0] + a[1] * b[1] + a[2] * b[2] + a[3] * b[3];
}

// Mpack[q] with q = (ki*4 + jt)*32 + lane holds (M[k0][j], M[k0+1][j])
// where k0 = 4*ki + 2*(lane>=16), j = jt*16 + (lane&15)  — the exact per-lane
// B-fragment layout of V_WMMA_F32_16X16X4_F32.
__global__ void ffm_build_Mpack(const float* __restrict__ V, float* __restrict__ Mw) {
    int q = blockIdx.x * blockDim.x + threadIdx.x;
    if (q >= MPACK_N) return;
    int lane = q & 31;
    int jt   = (q >> 5) & 3;
    int ki   = q >> 7;
    int k0   = 4 * ki + ((lane >> 4) << 1);
    int j    = jt * 16 + (lane & 15);
    Mw[2 * q + 0] = mval(V, k0,     j);
    Mw[2 * q + 1] = mval(V, k0 + 1, j);
}

__global__ __launch_bounds__(NTHREADS)
void ffm_main(const float* __restrict__ x, const float* __restrict__ W,
              const float* __restrict__ bias, const float* __restrict__ Mg,
              float* __restrict__ out) {
    extern __shared__ float smem[];
    float* Mp    = smem;                        // 4096 floats (packed B frags)
    float* Xe    = smem + 2 * MPACK_N;          // 12 * EPLANE floats
    float* lin   = Xe + EM * EPLANE;            // 16
    float* inter = lin + ROWS;                  // 16

    const int tid  = threadIdx.x;
    const int lane = tid & 31;
    const int e    = tid >> 5;                  // wave id == embedding lane

    if (tid < ROWS) { lin[tid] = bias[0]; inter[tid] = 0.f; }

    // ---- stage packed M (linear float4 copy) ----
    {
        const float4* mg4 = reinterpret_cast<const float4*>(Mg);
        float4* mp4 = reinterpret_cast<float4*>(Mp);
        for (int f = tid; f < 2 * MPACK_N / 4; f += NTHREADS)
            mp4[f] = mg4[f];
    }

    // ---- stage x tile into e-major planes + fused linear part ----
    {
        const int row = tid / 24;               // 16 rows, 24 threads each
        const int t24 = tid % 24;
        const float4* xg = reinterpret_cast<const float4*>(
            x + (size_t)blockIdx.x * ROWS * DIM) + row * (DIM / 4);
        const float4* wg = reinterpret_cast<const float4*>(W);
        float lin_s = 0.f;
        #pragma unroll
        for (int q = 0; q < 8; ++q) {
            const int c = t24 * 8 + q;          // float4 column 0..191
            float4 v = xg[c];
            float4 w = wg[c];
            lin_s += v.x * w.x + v.y * w.y + v.z * w.z + v.w * w.w;
            // d0 = 4c = i*12 + e0 with e0 = 4*(c%3) in {0,4,8}; no e-wrap.
            const int c3 = c / 3;
            const int e0 = (c - 3 * c3) * 4;
            const int i  = c3 & 63;             // c3 == i (0..63); row offset is in xg
            float* dst = &Xe[e0 * EPLANE + row * XSR + i];
            dst[0 * EPLANE] = v.x;
            dst[1 * EPLANE] = v.y;
            dst[2 * EPLANE] = v.z;
            dst[3 * EPLANE] = v.w;
        }
        atomicAdd(&lin[row], lin_s);
    }
    __syncthreads();

    // ---- interaction: T[:, j, e] = sum_i X[:, i, e] * M[i, j] via WMMA ----
    const int ar = lane & 15;                   // A row / B col-within-tile
    const int kh = (lane >> 4) << 1;            // lanes 16..31 hold K=2,3

    const float* Ab = &Xe[e * EPLANE + ar * XSR + kh];   // + 4*ki -> v2f
    const float* Bb = &Mp[lane * 2];                     // + (ki*4+jt)*64 -> v2f

    v8f acc0 = {}, acc1 = {}, acc2 = {}, acc3 = {};
    #pragma unroll
    for (int ki = 0; ki < 16; ++ki) {           // K = 64 in steps of 4
        v2f a  = *reinterpret_cast<const v2f*>(Ab + 4 * ki);
        v2f b0 = *reinterpret_cast<const v2f*>(Bb + (ki * 4 + 0) * 64);
        v2f b1 = *reinterpret_cast<const v2f*>(Bb + (ki * 4 + 1) * 64);
        v2f b2 = *reinterpret_cast<const v2f*>(Bb + (ki * 4 + 2) * 64);
        v2f b3 = *reinterpret_cast<const v2f*>(Bb + (ki * 4 + 3) * 64);
        acc0 = __builtin_amdgcn_wmma_f32_16x16x4_f32(false, a, false, b0,
                                                     (short)0, acc0, false, false);
        acc1 = __builtin_amdgcn_wmma_f32_16x16x4_f32(false, a, false, b1,
                                                     (short)0, acc1, false, false);
        acc2 = __builtin_amdgcn_wmma_f32_16x16x4_f32(false, a, false, b2,
                                                     (short)0, acc2, false, false);
        acc3 = __builtin_amdgcn_wmma_f32_16x16x4_f32(false, a, false, b3,
                                                     (short)0, acc3, false, false);
    }

    // ---- fold: sum_j T[row, j, e] * X[row, j, e] ----
    // D layout: vgpr g, lanes 0-15 -> row g, lanes 16-31 -> row g+8; col = ar.
    const int rbase = (lane >> 4) << 3;
    const float* Fb = &Xe[e * EPLANE + rbase * XSR + ar];
    float partial[8];
    #pragma unroll
    for (int g = 0; g < 8; ++g) {
        float p;
        p  = acc0[g] * Fb[g * XSR +  0];
        p += acc1[g] * Fb[g * XSR + 16];
        p += acc2[g] * Fb[g * XSR + 32];
        p += acc3[g] * Fb[g * XSR + 48];
        partial[g] = p;
    }
    // reduce across the 16 lanes of each half (wave32 xor shuffles)
    #pragma unroll
    for (int g = 0; g < 8; ++g) {
        float v = partial[g];
        v += __shfl_xor(v, 1, 32);
        v += __shfl_xor(v, 2, 32);
        v += __shfl_xor(v, 4, 32);
        v += __shfl_xor(v, 8, 32);
        partial[g] = v;
    }
    if ((lane & 15) == 0) {
        #pragma unroll
        for (int g = 0; g < 8; ++g)
            atomicAdd(&inter[rbase + g], partial[g]);
    }
    __syncthreads();

    // ---- sigmoid + store ----
    if (tid < ROWS) {
        float z = lin[tid] + inter[tid];
        out[(size_t)blockIdx.x * ROWS + tid] = 1.f / (1.f + __expf(-z));
    }
}

extern "C" void kernel_launch(void* const* d_in, const int* in_sizes, int n_in,
                              void* d_out, int out_size, void* d_ws, size_t ws_size,
                              hipStream_t stream) {
    const float* x  = (const float*)d_in[0];   // [32768, 768]
    const float* W  = (const float*)d_in[1];   // [1, 768]
    const float* b  = (const float*)d_in[2];   // [1]
    const float* V  = (const float*)d_in[3];   // [64, 64, 4]
    float* out = (float*)d_out;                // [32768]
    float* Mw  = (float*)d_ws;                 // 4096 floats: packed B fragments

    ffm_build_Mpack<<<(MPACK_N + 255) / 256, 256, 0, stream>>>(V, Mw);

    size_t shmem = (size_t)(2 * MPACK_N + EM * EPLANE + 2 * ROWS) * sizeof(float);
    ffm_main<<<B_TOTAL / ROWS, NTHREADS, shmem, stream>>>(x, W, b, Mw, out);
}